// RigidBodySimulator_13804024889455
// MI455X (gfx1250) — compile-verified
//
#include <hip/hip_runtime.h>
#include <cstdint>
#include <cstddef>

// ---------------------------------------------------------------------------
// Rigid-body contact step, MI455X (gfx1250).
// Heavy part: masked reduction over 8.4M float3 vertices (~100 MB streamed,
// memory-bound: ~4.3us at 23.3 TB/s). Tail: O(1) 3x3 physics on 1 thread.
// ---------------------------------------------------------------------------

typedef float fvec4 __attribute__((ext_vector_type(4)));
typedef int   v4i_vs __attribute__((vector_size(16)));   // int4 (vector_size)

#define NB    2048          // reduction blocks (partials rows in d_ws)
#define BT    256           // threads per block (8 wave32)
#define DT_D  (1.0/600.0)   // 1/60/10

// ---- CDNA5 async global->LDS path (guarded; falls back to NT loads) -------
#if defined(__HIP_DEVICE_COMPILE__)
#  if __has_builtin(__builtin_amdgcn_global_load_async_to_lds_b128) && \
      __has_builtin(__builtin_amdgcn_s_wait_asynccnt)
#    define USE_ASYNC 1
#  else
#    define USE_ASYNC 0
#  endif
#else
#  define USE_ASYNC 0
#endif

#if USE_ASYNC
// Signature (from clang diagnostics): param0 is `v4i*` (generic), 4 args total.
// param1 assumed `addrspace(3) v4i*` (LDS destination), then imm offset, imm cpol.
#define GLD_ASYNC_B128(gp, lp)                                          \
  __builtin_amdgcn_global_load_async_to_lds_b128(                       \
      (v4i_vs*)(gp),                                                    \
      (__attribute__((address_space(3))) v4i_vs*)(lp), 0, 0)
#endif

// per-vertex accumulate: d = r10*x + r11*y + r12*z + cpl ; mask = d < 0
#define ACC_VTX(px, py, pz)                                             \
  do {                                                                  \
    float _d = fmaf(r10, (px), fmaf(r11, (py), fmaf(r12, (pz), cpl)));  \
    bool _h = _d < 0.0f;                                                \
    cnt += _h ? 1.0f : 0.0f;                                            \
    sx  += _h ? (px) : 0.0f;                                            \
    sy  += _h ? (py) : 0.0f;                                            \
    sz  += _h ? (pz) : 0.0f;                                            \
  } while (0)

// ---------------------------------------------------------------------------
// Kernel 1: streaming masked reduction. Each "group" = 4 vertices = 3 float4.
// Writes one (cnt, sx, sy, sz) row per block to partials (deterministic).
// ---------------------------------------------------------------------------
__global__ __launch_bounds__(BT) void rb_partial_reduce(
    const fvec4* __restrict__ X4,       // 3N/4 float4s
    const float* __restrict__ mc,       // 3
    const float* __restrict__ tr,       // 3
    const float* __restrict__ qu,       // 4
    float* __restrict__ partials,       // gridDim.x * 4
    int numGroups)
{
  // Uniform: row 1 of R(q) and the plane constant.
  float qw = qu[0], qx = qu[1], qy = qu[2], qz = qu[3];
  float inq = 1.0f / sqrtf(qw*qw + qx*qx + qy*qy + qz*qz);
  qw *= inq; qx *= inq; qy *= inq; qz *= inq;
  const float r10 = 2.0f * (qx*qy + qw*qz);
  const float r11 = 1.0f - 2.0f * (qx*qx + qz*qz);
  const float r12 = 2.0f * (qy*qz - qw*qx);
  const float cpl = tr[1] + mc[1] - (r10*mc[0] + r11*mc[1] + r12*mc[2]);

  const int tid = threadIdx.x;
  float cnt = 0.0f, sx = 0.0f, sy = 0.0f, sz = 0.0f;

#if USE_ASYNC
  // Double-buffered async copy: tile = BT groups = 3*BT float4 = 12 KB.
  __shared__ fvec4 tile[2][3 * BT];
  const int numTiles = numGroups / BT;
  int t = (int)blockIdx.x;
  int buf = 0;
  if (t < numTiles) {
    {
      const fvec4* g = X4 + (size_t)t * (3 * BT);
      GLD_ASYNC_B128(g + tid,          &tile[0][tid]);
      GLD_ASYNC_B128(g + BT + tid,     &tile[0][BT + tid]);
      GLD_ASYNC_B128(g + 2*BT + tid,   &tile[0][2*BT + tid]);
    }
    while (t < numTiles) {
      const int tn = t + (int)gridDim.x;
      if (tn < numTiles) {
        const fvec4* g = X4 + (size_t)tn * (3 * BT);
        const int nxt = buf ^ 1;
        GLD_ASYNC_B128(g + tid,          &tile[nxt][tid]);
        GLD_ASYNC_B128(g + BT + tid,     &tile[nxt][BT + tid]);
        GLD_ASYNC_B128(g + 2*BT + tid,   &tile[nxt][2*BT + tid]);
        __builtin_amdgcn_s_wait_asynccnt(3);   // oldest 3 (current tile) done
      } else {
        __builtin_amdgcn_s_wait_asynccnt(0);
      }
      __syncthreads();                         // all waves' copies landed
      const fvec4 a  = tile[buf][3*tid + 0];   // x0 y0 z0 x1
      const fvec4 b  = tile[buf][3*tid + 1];   // y1 z1 x2 y2
      const fvec4 c4 = tile[buf][3*tid + 2];   // z2 x3 y3 z3
      ACC_VTX(a.x,  a.y,  a.z);
      ACC_VTX(a.w,  b.x,  b.y);
      ACC_VTX(b.z,  b.w,  c4.x);
      ACC_VTX(c4.y, c4.z, c4.w);
      __syncthreads();                         // buffer free for reuse
      buf ^= 1;
      t = tn;
    }
  }
  // Remainder groups (none for N=8388608) via direct loads.
  for (int g = numTiles * BT + (int)blockIdx.x * BT + tid; g < numGroups;
       g += (int)gridDim.x * BT) {
    const size_t base = 3 * (size_t)g;
    const fvec4 a  = X4[base + 0];
    const fvec4 b  = X4[base + 1];
    const fvec4 c4 = X4[base + 2];
    ACC_VTX(a.x,  a.y,  a.z);
    ACC_VTX(a.w,  b.x,  b.y);
    ACC_VTX(b.z,  b.w,  c4.x);
    ACC_VTX(c4.y, c4.z, c4.w);
  }
#else
  // Fallback: grid-stride nontemporal b128 loads (stream-once, spare L2).
  for (int g = (int)blockIdx.x * BT + tid; g < numGroups;
       g += (int)gridDim.x * BT) {
    const size_t base = 3 * (size_t)g;
    const fvec4 a  = __builtin_nontemporal_load(X4 + base + 0);
    const fvec4 b  = __builtin_nontemporal_load(X4 + base + 1);
    const fvec4 c4 = __builtin_nontemporal_load(X4 + base + 2);
    ACC_VTX(a.x,  a.y,  a.z);
    ACC_VTX(a.w,  b.x,  b.y);
    ACC_VTX(b.z,  b.w,  c4.x);
    ACC_VTX(c4.y, c4.z, c4.w);
  }
#endif

  // wave32 butterfly reduce (deterministic order)
  for (int off = 16; off > 0; off >>= 1) {
    cnt += __shfl_xor(cnt, off, 32);
    sx  += __shfl_xor(sx,  off, 32);
    sy  += __shfl_xor(sy,  off, 32);
    sz  += __shfl_xor(sz,  off, 32);
  }
  __shared__ float wred[BT / 32][4];
  const int wid = tid >> 5, lane = tid & 31;
  if (lane == 0) {
    wred[wid][0] = cnt; wred[wid][1] = sx; wred[wid][2] = sy; wred[wid][3] = sz;
  }
  __syncthreads();
  if (tid == 0) {
    float c = 0.f, a = 0.f, b = 0.f, d = 0.f;
    for (int w = 0; w < BT / 32; ++w) {
      c += wred[w][0]; a += wred[w][1]; b += wred[w][2]; d += wred[w][3];
    }
    partials[blockIdx.x * 4 + 0] = c;
    partials[blockIdx.x * 4 + 1] = a;
    partials[blockIdx.x * 4 + 2] = b;
    partials[blockIdx.x * 4 + 3] = d;
  }
}

// ---------------------------------------------------------------------------
// small double-precision 3x3 helpers
// ---------------------------------------------------------------------------
__device__ __forceinline__ void m3mul(const double* A, const double* B, double* C) {
  for (int r = 0; r < 3; ++r)
    for (int c = 0; c < 3; ++c)
      C[3*r + c] = A[3*r]*B[c] + A[3*r+1]*B[3+c] + A[3*r+2]*B[6+c];
}
__device__ __forceinline__ void m3mulBT(const double* A, const double* B, double* C) {
  // C = A * B^T
  for (int r = 0; r < 3; ++r)
    for (int c = 0; c < 3; ++c)
      C[3*r + c] = A[3*r]*B[3*c] + A[3*r+1]*B[3*c+1] + A[3*r+2]*B[3*c+2];
}
__device__ __forceinline__ void m3vec(const double* A, const double* x, double* y) {
  for (int r = 0; r < 3; ++r)
    y[r] = A[3*r]*x[0] + A[3*r+1]*x[1] + A[3*r+2]*x[2];
}
__device__ __forceinline__ void m3inv(const double* A, double* X) {
  const double c00 = A[4]*A[8] - A[5]*A[7];
  const double c01 = A[5]*A[6] - A[3]*A[8];
  const double c02 = A[3]*A[7] - A[4]*A[6];
  const double id  = 1.0 / (A[0]*c00 + A[1]*c01 + A[2]*c02);
  X[0] = c00*id;                  X[1] = (A[2]*A[7]-A[1]*A[8])*id; X[2] = (A[1]*A[5]-A[2]*A[4])*id;
  X[3] = c01*id;                  X[4] = (A[0]*A[8]-A[2]*A[6])*id; X[5] = (A[2]*A[3]-A[0]*A[5])*id;
  X[6] = c02*id;                  X[7] = (A[1]*A[6]-A[0]*A[7])*id; X[8] = (A[0]*A[4]-A[1]*A[3])*id;
}

// ---------------------------------------------------------------------------
// Kernel 2: final deterministic reduce + O(1) physics tail -> 7 outputs.
// ---------------------------------------------------------------------------
__global__ __launch_bounds__(256) void rb_finalize(
    const float* __restrict__ partials, int nparts,
    const float* __restrict__ mc, const float* __restrict__ tr,
    const float* __restrict__ qu, const float* __restrict__ vv,
    const float* __restrict__ om, const float* __restrict__ kn,
    const float* __restrict__ mu, const float* __restrict__ ld,
    const float* __restrict__ ad, const float* __restrict__ Iref,
    float* __restrict__ out, int Nverts)
{
  __shared__ double red[256][4];
  double a0 = 0, a1 = 0, a2 = 0, a3 = 0;
  for (int i = threadIdx.x; i < nparts; i += 256) {
    a0 += (double)partials[4*i + 0];
    a1 += (double)partials[4*i + 1];
    a2 += (double)partials[4*i + 2];
    a3 += (double)partials[4*i + 3];
  }
  red[threadIdx.x][0] = a0; red[threadIdx.x][1] = a1;
  red[threadIdx.x][2] = a2; red[threadIdx.x][3] = a3;
  __syncthreads();
  for (int s = 128; s > 0; s >>= 1) {
    if ((int)threadIdx.x < s)
      for (int j = 0; j < 4; ++j)
        red[threadIdx.x][j] += red[threadIdx.x + s][j];
    __syncthreads();
  }
  if (threadIdx.x != 0) return;

  const double cnt = red[0][0];
  const double s3[3] = { red[0][1], red[0][2], red[0][3] };

  // R from normalized quaternion (raw q kept for the integration step)
  const double q0 = qu[0], q1 = qu[1], q2 = qu[2], q3 = qu[3];
  const double qn = sqrt(q0*q0 + q1*q1 + q2*q2 + q3*q3);
  const double w = q0/qn, x = q1/qn, y = q2/qn, z = q3/qn;
  const double R[9] = {
    1 - 2*(y*y + z*z), 2*(x*y - w*z),     2*(x*z + w*y),
    2*(x*y + w*z),     1 - 2*(x*x + z*z), 2*(y*z - w*x),
    2*(x*z - w*y),     2*(y*z + w*x),     1 - 2*(x*x + y*y) };

  const double mcd[3] = { mc[0], mc[1], mc[2] };
  const double trd[3] = { tr[0], tr[1], tr[2] };
  double Rmc[3], Rs[3];
  m3vec(R, mcd, Rmc);
  m3vec(R, s3, Rs);
  // sum_position = cnt*(t + mc - R*mc) + R*sum(mask*x)
  double sum_pos[3];
  for (int i = 0; i < 3; ++i)
    sum_pos[i] = cnt * (trd[i] + mcd[i] - Rmc[i]) + Rs[i];

  const double denom = cnt > 1.0 ? cnt : 1.0;
  const double ri[3] = { sum_pos[0]/denom, sum_pos[1]/denom, sum_pos[2]/denom };
  double Ri[3];
  m3vec(R, ri, Ri);

  const double vd[3]  = { vv[0], vv[1], vv[2] };
  const double omd[3] = { om[0], om[1], om[2] };
  const double vi[3] = {
    vd[0] + (omd[1]*Ri[2] - omd[2]*Ri[1]),
    vd[1] + (omd[2]*Ri[0] - omd[0]*Ri[2]),
    vd[2] + (omd[0]*Ri[1] - omd[1]*Ri[0]) };

  const double knd = kn[0], mud = mu[0];
  const double vtn = sqrt(vi[0]*vi[0] + vi[2]*vi[2]);   // |v_i_t|
  const double vnn = fabs(vi[1]);                       // |v_i_n|
  double alpha = 1.0 - mud * (1.0 + knd) * (vtn / (vnn + 1e-10));
  if (alpha < 0.0) alpha = 0.0;
  // vi_new - vi
  const double rhs[3] = { (alpha - 1.0)*vi[0], (-knd - 1.0)*vi[1],
                          (alpha - 1.0)*vi[2] };

  double Irefd[9];
  for (int i = 0; i < 9; ++i) Irefd[i] = Iref[i];
  double T1[9], Iw[9], Iinv[9];
  m3mul(R, Irefd, T1);
  m3mulBT(T1, R, Iw);          // R * Iref * R^T
  m3inv(Iw, Iinv);

  const double Rri[9] = { 0, -Ri[2], Ri[1],  Ri[2], 0, -Ri[0],  -Ri[1], Ri[0], 0 };
  double T2[9], T3[9];
  m3mul(Rri, Iinv, T2);
  m3mul(T2, Rri, T3);

  const double invm = 1.0 / (double)Nverts;
  double K[9];
  for (int i = 0; i < 9; ++i) K[i] = -T3[i];
  K[0] += invm; K[4] += invm; K[8] += invm;
  double Kinv[9], J[3];
  m3inv(K, Kinv);
  m3vec(Kinv, rhs, J);

  double vout[3] = { vd[0], vd[1] - 9.8 * DT_D * (double)ld[0], vd[2] };
  double oout[3] = { omd[0]*(double)ad[0], omd[1]*(double)ad[0], omd[2]*(double)ad[0] };
  if (cnt > 0.0) {
    double RJ[3], dI[3];
    m3vec(Rri, J, RJ);
    m3vec(Iinv, RJ, dI);
    for (int i = 0; i < 3; ++i) { vout[i] += J[i] * invm; oout[i] += dI[i]; }
  }

  const double wt0 = oout[0]*DT_D*0.5, wt1 = oout[1]*DT_D*0.5, wt2 = oout[2]*DT_D*0.5;
  // dq = quat_mul([0, wt], q_raw)
  const double dqw = -(wt0*q1 + wt1*q2 + wt2*q3);
  const double dqx = wt0*q0 + wt1*q3 - wt2*q2;
  const double dqy = wt1*q0 + wt2*q1 - wt0*q3;
  const double dqz = wt2*q0 + wt0*q2 - wt1*q1;

  const double nq0 = q0 + dqw, nq1 = q1 + dqx, nq2 = q2 + dqy, nq3 = q3 + dqz;
  const double nn = sqrt(nq0*nq0 + nq1*nq1 + nq2*nq2 + nq3*nq3);

  out[0] = (float)(trd[0] + DT_D * vout[0]);
  out[1] = (float)(trd[1] + DT_D * vout[1]);
  out[2] = (float)(trd[2] + DT_D * vout[2]);
  out[3] = (float)(nq0 / nn);
  out[4] = (float)(nq1 / nn);
  out[5] = (float)(nq2 / nn);
  out[6] = (float)(nq3 / nn);
}

// ---------------------------------------------------------------------------
extern "C" void kernel_launch(void* const* d_in, const int* in_sizes, int n_in,
                              void* d_out, int out_size, void* d_ws, size_t ws_size,
                              hipStream_t stream) {
  const float* x_  = (const float*)d_in[0];   // (N,3)
  const float* mc  = (const float*)d_in[1];   // (3,)
  const float* tr  = (const float*)d_in[2];   // (3,)
  const float* qu  = (const float*)d_in[3];   // (4,)
  const float* vv  = (const float*)d_in[4];   // (3,)
  const float* om  = (const float*)d_in[5];   // (3,)
  const float* kn  = (const float*)d_in[6];   // (1,)
  const float* mu  = (const float*)d_in[7];   // (1,)
  const float* ld  = (const float*)d_in[8];   // (1,)
  const float* ad  = (const float*)d_in[9];   // (1,)
  const float* Ir  = (const float*)d_in[10];  // (3,3)
  float* out = (float*)d_out;
  float* partials = (float*)d_ws;

  const int Nverts = in_sizes[0] / 3;
  const int numGroups = Nverts / 4;           // 4 vertices per thread-group

  int nb = NB;
  if ((size_t)nb * 16 > ws_size) nb = (int)(ws_size / 16);
  if (nb < 1) nb = 1;

  rb_partial_reduce<<<nb, BT, 0, stream>>>(
      (const fvec4*)x_, mc, tr, qu, partials, numGroups);
  rb_finalize<<<1, 256, 0, stream>>>(
      partials, nb, mc, tr, qu, vv, om, kn, mu, ld, ad, Ir, out, Nverts);
}